// GCN_89163521065178
// MI455X (gfx1250) — compile-verified
//
#include <hip/hip_runtime.h>
#include <hip/hip_bf16.h>

#define NN      50000
#define NE      600000
#define NG      64
#define NLAYERS 4
#define DD      128
#define BN_EPS  1e-5f

typedef __attribute__((ext_vector_type(2))) float v2f;
typedef __attribute__((ext_vector_type(8))) float v8f;

// ---------------- embedding gather: x[n][:] = emb[x_node[n]][:] ----------------
__global__ void k_embed(const int* __restrict__ xn, const float* __restrict__ emb,
                        float* __restrict__ x) {
    int i = blockIdx.x * blockDim.x + threadIdx.x;      // over NN*32 float4 slots
    if (i >= NN * 32) return;
    int n = i >> 5, d4 = i & 31;
    ((float4*)x)[(size_t)n * 32 + d4] =
        ((const float4*)emb)[(size_t)xn[n] * 32 + d4];
}

// ---------------- copy (agg := x, the (1+eps)*x self term) ----------------
__global__ void k_copy(const float* __restrict__ src, float* __restrict__ dst, int n4) {
    int i = blockIdx.x * blockDim.x + threadIdx.x;
    if (i < n4) ((float4*)dst)[i] = ((const float4*)src)[i];
}

// ---------------- edge scatter-add: agg[dst] += x[src]  (wave32 per edge) ----------------
__global__ void k_scatter(const int* __restrict__ ei, const float* __restrict__ x,
                          float* __restrict__ agg) {
    int t = blockIdx.x * blockDim.x + threadIdx.x;      // NE*32 threads
    int e = t >> 5, lane = t & 31;
    if (e >= NE) return;
    int s = ei[e];
    int d = ei[NE + e];
    float4 v = ((const float4*)(x + (size_t)s * DD))[lane];
    float* dp = agg + (size_t)d * DD + lane * 4;
    atomicAdd(dp + 0, v.x);
    atomicAdd(dp + 1, v.y);
    atomicAdd(dp + 2, v.z);
    atomicAdd(dp + 3, v.w);
}

// ---------------- zero small scratch ----------------
__global__ void k_zero(float* __restrict__ p, int n) {
    int i = blockIdx.x * blockDim.x + threadIdx.x;
    if (i < n) p[i] = 0.f;
}

// ---------------- GEMM: T = X @ W + bias, fused BN-stat accumulation ----------------
// 256 threads = 8 waves. Block owns 16 node rows; wave w owns output cols [16w,16w+16).
// K-loop: 32 x V_WMMA_F32_16X16X4_F32 (full fp32 precision, matches reference).
__global__ void k_gemm(const float* __restrict__ X, const float* __restrict__ W,
                       const float* __restrict__ bias, float* __restrict__ T,
                       float* __restrict__ colsum, float* __restrict__ colsq) {
    __shared__ float lds[16 * 132];                     // 16 rows, padded stride 132
    const int rowbase = blockIdx.x * 16;
    const int t = threadIdx.x;

    // cooperative stage of 16x128 A-panel into LDS (512 float4 loads / 256 threads)
    for (int i = t; i < 16 * 32; i += 256) {
        int r = i >> 5, c4 = i & 31;
        float4 v = ((const float4*)(X + (size_t)(rowbase + r) * DD))[c4];
        float* dst = &lds[r * 132 + c4 * 4];
        dst[0] = v.x; dst[1] = v.y; dst[2] = v.z; dst[3] = v.w;
    }
    __syncthreads();

    const int wave = t >> 5, lane = t & 31;
    const int col  = wave * 16 + (lane & 15);           // N index (B/C/D lane slot)
    const int koff = (lane >> 4) << 1;                  // lanes 16-31 carry K+2,K+3
    const int arow = lane & 15;                         // M index (A lane slot)

    v8f c;
    {
        float bv = bias[col];
        for (int v = 0; v < 8; ++v) c[v] = bv;          // fold bias into accumulator
    }
    const float* ap = &lds[arow * 132];
    for (int k = 0; k < DD; k += 4) {
        v2f a, b;
        a.x = ap[k + koff];
        a.y = ap[k + koff + 1];
        b.x = W[(size_t)(k + koff)     * DD + col];
        b.y = W[(size_t)(k + koff + 1) * DD + col];
        c = __builtin_amdgcn_wmma_f32_16x16x4_f32(false, a, false, b,
                                                  (short)0, c, false, false);
    }

    // D layout: VGPR v -> row v + 8*(lane>=16), col = lane&15
    const int rtop = (lane >> 4) << 3;
    float s = 0.f, sq = 0.f;
    for (int v = 0; v < 8; ++v) {
        float d = c[v];
        T[(size_t)(rowbase + rtop + v) * DD + col] = d;
        s  += d;
        sq += d * d;
    }
    atomicAdd(&colsum[col], s);
    atomicAdd(&colsq[col], sq);
}

// ---------------- BN (from fused stats) + ReLU ----------------
__global__ void k_bn(const float* __restrict__ T, const float* __restrict__ colsum,
                     const float* __restrict__ colsq, const float* __restrict__ g,
                     const float* __restrict__ b, float* __restrict__ out) {
    int i = blockIdx.x * blockDim.x + threadIdx.x;      // NN*DD
    if (i >= NN * DD) return;
    int col = i & (DD - 1);
    const float invn = 1.f / (float)NN;
    float mu  = colsum[col] * invn;
    float var = colsq[col] * invn - mu * mu;
    float v = (T[i] - mu) * rsqrtf(var + BN_EPS) * g[col] + b[col];
    out[i] = fmaxf(v, 0.f);
}

// ---------------- per-graph mean-pool accumulation ----------------
__global__ void k_pool(const float* __restrict__ x, const int* __restrict__ batch,
                       float* __restrict__ pooled, float* __restrict__ counts) {
    int t = blockIdx.x * blockDim.x + threadIdx.x;      // NN*32
    int n = t >> 5, lane = t & 31;
    if (n >= NN) return;
    int gidx = batch[n];
    float4 v = ((const float4*)(x + (size_t)n * DD))[lane];
    float* dp = pooled + (size_t)gidx * DD + lane * 4;
    atomicAdd(dp + 0, v.x);
    atomicAdd(dp + 1, v.y);
    atomicAdd(dp + 2, v.z);
    atomicAdd(dp + 3, v.w);
    if (lane == 0) atomicAdd(&counts[gidx], 1.f);
}

// ---------------- head MLP: one block per graph ----------------
__global__ void k_mlp(const float* __restrict__ pooled, const float* __restrict__ counts,
                      const float* __restrict__ w1, const float* __restrict__ b1,
                      const float* __restrict__ w2, const float* __restrict__ b2,
                      float* __restrict__ out) {
    __shared__ float pm[DD];
    __shared__ float h1[2 * DD];
    int g = blockIdx.x, t = threadIdx.x;                // 256 threads
    float inv = 1.f / fmaxf(counts[g], 1.f);
    if (t < DD) pm[t] = pooled[(size_t)g * DD + t] * inv;
    __syncthreads();
    float acc = b1[t];
    for (int k = 0; k < DD; ++k) acc += pm[k] * w1[(size_t)k * 2 * DD + t];
    h1[t] = fmaxf(acc, 0.f);
    __syncthreads();
    if (t < 10) {
        float a2 = b2[t];
        for (int k = 0; k < 2 * DD; ++k) a2 += h1[k] * w2[(size_t)k * 10 + t];
        out[(size_t)g * 10 + t] = a2;
    }
}

extern "C" void kernel_launch(void* const* d_in, const int* in_sizes, int n_in,
                              void* d_out, int out_size, void* d_ws, size_t ws_size,
                              hipStream_t stream) {
    const int*   x_node  = (const int*)d_in[0];
    const int*   ei      = (const int*)d_in[1];
    const int*   batch   = (const int*)d_in[2];
    const float* emb     = (const float*)d_in[3];
    const float* conv_w1 = (const float*)d_in[4];
    const float* conv_b1 = (const float*)d_in[5];
    const float* bn1_g   = (const float*)d_in[6];
    const float* bn1_b   = (const float*)d_in[7];
    const float* conv_w2 = (const float*)d_in[8];
    const float* conv_b2 = (const float*)d_in[9];
    const float* bn2_g   = (const float*)d_in[10];
    const float* bn2_b   = (const float*)d_in[11];
    const float* lin1_w  = (const float*)d_in[12];
    const float* lin1_b  = (const float*)d_in[13];
    const float* lin2_w  = (const float*)d_in[14];
    const float* lin2_b  = (const float*)d_in[15];
    float* out = (float*)d_out;

    const size_t NND = (size_t)NN * DD;                 // 6.4M floats
    float* bufX   = (float*)d_ws;                       // current node features
    float* bufH   = bufX + NND;                         // agg / post-BN1
    float* bufT   = bufH + NND;                         // raw GEMM output
    float* colsum = bufT + NND;                         // 128
    float* colsq  = colsum + DD;                        // 128
    float* pooled = colsq + DD;                         // 64*128
    float* counts = pooled + (size_t)NG * DD;           // 64

    const int B = 256;
    const int g_vec4 = (NN * 32) / B;                   // 6250, exact
    const int g_elem = (NN * DD) / B;                   // 25000, exact
    const int g_edge = (NE * 32) / B;                   // 75000, exact
    const int g_gemm = NN / 16;                         // 3125, exact

    k_embed<<<g_vec4, B, 0, stream>>>(x_node, emb, bufX);

    for (int l = 0; l < NLAYERS; ++l) {
        const float* w1 = conv_w1 + (size_t)l * DD * DD;
        const float* w2 = conv_w2 + (size_t)l * DD * DD;

        // agg = x, then agg[dst] += x[src]
        k_copy<<<g_vec4, B, 0, stream>>>(bufX, bufH, NN * DD / 4);
        k_scatter<<<g_edge, B, 0, stream>>>(ei, bufX, bufH);

        // h = relu(bn(agg @ W1 + b1))
        k_zero<<<1, B, 0, stream>>>(colsum, 2 * DD);
        k_gemm<<<g_gemm, B, 0, stream>>>(bufH, w1, conv_b1 + l * DD, bufT, colsum, colsq);
        k_bn<<<g_elem, B, 0, stream>>>(bufT, colsum, colsq,
                                       bn1_g + l * DD, bn1_b + l * DD, bufH);

        // x = relu(bn(h @ W2 + b2))
        k_zero<<<1, B, 0, stream>>>(colsum, 2 * DD);
        k_gemm<<<g_gemm, B, 0, stream>>>(bufH, w2, conv_b2 + l * DD, bufT, colsum, colsq);
        k_bn<<<g_elem, B, 0, stream>>>(bufT, colsum, colsq,
                                       bn2_g + l * DD, bn2_b + l * DD, bufX);
    }

    k_zero<<<(NG * DD + NG + B - 1) / B, B, 0, stream>>>(pooled, NG * DD + NG);
    k_pool<<<g_vec4, B, 0, stream>>>(bufX, batch, pooled, counts);
    k_mlp<<<NG, B, 0, stream>>>(pooled, counts, lin1_w, lin1_b, lin2_w, lin2_b, out);
}